// GNN_30064771072959
// MI455X (gfx1250) — compile-verified
//
#include <hip/hip_runtime.h>
#include <hip/hip_bf16.h>

typedef __attribute__((ext_vector_type(2))) float v2f;
typedef __attribute__((ext_vector_type(8))) float v8f;

#define LDSTRIDE 132   // 16 rows * 132 dwords; 132 % 64 == 4 -> rows spread across banks

// ---------------------------------------------------------------------------
// Degree accumulation: deg_out[src[e]] += 1, deg_in[dst[e]] += 1
// ---------------------------------------------------------------------------
__global__ void gnn_deg_kernel(const int* __restrict__ src,
                               const int* __restrict__ dst,
                               float* __restrict__ deg_out,
                               float* __restrict__ deg_in, int E) {
    int e = blockIdx.x * blockDim.x + threadIdx.x;
    if (e < E) {
        atomicAdd(&deg_out[src[e]], 1.0f);
        atomicAdd(&deg_in[dst[e]], 1.0f);
    }
}

// In-place: d[i] = 1/sqrt(max(d[i],1))   (applied to [nsrc | ndst], length 2N)
__global__ void gnn_norm_kernel(float* __restrict__ d, int n) {
    int i = blockIdx.x * blockDim.x + threadIdx.x;
    if (i < n) {
        float v = fmaxf(d[i], 1.0f);
        d[i] = 1.0f / sqrtf(v);
    }
}

// ---------------------------------------------------------------------------
// X = (H * nsrc[:,None]) @ W   with fp32 WMMA 16x16x4.
// Block: 16 rows x 128 cols. 128 threads = 4 waves; each wave -> 2 col tiles.
// A tile staged in LDS (scaled); B (W, 64KB) read from global (cache-hot).
// ---------------------------------------------------------------------------
__global__ __launch_bounds__(128) void gnn_gemm_kernel(
    const float* __restrict__ H, const float* __restrict__ nsrc,
    const float* __restrict__ W, float* __restrict__ X, int N) {
    __shared__ float As[16 * LDSTRIDE];

    const int row0 = blockIdx.x * 16;
    const int tid  = threadIdx.x;

    // Cooperative load of the 16x128 A tile, scaled by norm_src, float4 units.
    for (int i = tid; i < 16 * 32; i += 128) {
        int r  = i >> 5;
        int c4 = (i & 31) * 4;
        int row = row0 + r;
        int rowc = (row < N) ? row : (N - 1);          // clamp; extra rows unused
        float s = nsrc[rowc];
        float4 v = *(const float4*)(H + (size_t)rowc * 128 + c4);
        float4 sv = make_float4(v.x * s, v.y * s, v.z * s, v.w * s);
        *(float4*)(&As[r * LDSTRIDE + c4]) = sv;
    }
    __syncthreads();

    const int wave = tid >> 5;
    const int lane = tid & 31;
    const int ln   = lane & 15;                 // N-index within tile / M-row for A
    const int koff = (lane < 16) ? 0 : 2;       // A/B lane-half K offset (ISA layout)

    const int n0 = wave * 32;                   // first col tile of this wave
    const int n1 = n0 + 16;                     // second col tile

    v8f acc0 = {};
    v8f acc1 = {};

    for (int kb = 0; kb < 128; kb += 4) {
        // A fragment: lane<16 -> K=kb,kb+1 ; lane>=16 -> K=kb+2,kb+3 (row = ln)
        v2f a = *(const v2f*)(&As[ln * LDSTRIDE + kb + koff]);

        // B fragments from W (row-major 128x128): rows K=kb+koff, kb+koff+1
        const float* wp = W + (size_t)(kb + koff) * 128;
        v2f b0, b1;
        b0.x = wp[n0 + ln];
        b0.y = wp[128 + n0 + ln];
        b1.x = wp[n1 + ln];
        b1.y = wp[128 + n1 + ln];

        acc0 = __builtin_amdgcn_wmma_f32_16x16x4_f32(
            false, a, false, b0, (short)0, acc0, false, false);
        acc1 = __builtin_amdgcn_wmma_f32_16x16x4_f32(
            false, a, false, b1, (short)0, acc1, false, false);
    }

    // C/D layout: VGPR v -> M = v + (lane>=16 ? 8 : 0), N = lane&15
    const int mbase = row0 + ((lane < 16) ? 0 : 8);
#pragma unroll
    for (int v = 0; v < 8; ++v) {
        int row = mbase + v;
        if (row < N) {
            X[(size_t)row * 128 + n0 + ln] = acc0[v];
            X[(size_t)row * 128 + n1 + ln] = acc1[v];
        }
    }
}

// ---------------------------------------------------------------------------
// Edge scatter-add: AGG[dst[e], :] += X[src[e], :]
// 32 lanes per edge, float4 per lane (128 features).
// ---------------------------------------------------------------------------
__global__ __launch_bounds__(256) void gnn_agg_kernel(
    const float* __restrict__ X, const int* __restrict__ src,
    const int* __restrict__ dst, float* __restrict__ AGG, int E) {
    long long t = (long long)blockIdx.x * blockDim.x + threadIdx.x;
    int e = (int)(t >> 5);
    if (e >= E) return;
    int lane = (int)(t & 31);
    int s = src[e];
    int d = dst[e];
    float4 v = *(const float4*)(X + (size_t)s * 128 + lane * 4);
    float* out = AGG + (size_t)d * 128 + lane * 4;
    atomicAdd(out + 0, v.x);
    atomicAdd(out + 1, v.y);
    atomicAdd(out + 2, v.z);
    atomicAdd(out + 3, v.w);
}

// ---------------------------------------------------------------------------
// OUT[n,:] = relu?( AGG[n,:] * ndst[n] + bias )
// ---------------------------------------------------------------------------
__global__ __launch_bounds__(256) void gnn_fin_kernel(
    const float* __restrict__ AGG, const float* __restrict__ ndst,
    const float* __restrict__ bias, float* __restrict__ OUT, int N, int do_relu) {
    long long t = (long long)blockIdx.x * blockDim.x + threadIdx.x;
    if (t >= (long long)N * 32) return;
    int node = (int)(t >> 5);
    int c4   = (int)(t & 31) * 4;
    float s = ndst[node];
    float4 v = *(const float4*)(AGG + (size_t)node * 128 + c4);
    float4 b = *(const float4*)(bias + c4);
    float4 o;
    o.x = v.x * s + b.x;
    o.y = v.y * s + b.y;
    o.z = v.z * s + b.z;
    o.w = v.w * s + b.w;
    if (do_relu) {
        o.x = fmaxf(o.x, 0.0f);
        o.y = fmaxf(o.y, 0.0f);
        o.z = fmaxf(o.z, 0.0f);
        o.w = fmaxf(o.w, 0.0f);
    }
    *(float4*)(OUT + (size_t)node * 128 + c4) = o;
}

extern "C" void kernel_launch(void* const* d_in, const int* in_sizes, int n_in,
                              void* d_out, int out_size, void* d_ws, size_t ws_size,
                              hipStream_t stream) {
    const float* feature = (const float*)d_in[0];
    const int*   src     = (const int*)d_in[1];
    const int*   dst     = (const int*)d_in[2];
    const float* W1      = (const float*)d_in[3];
    const float* b1      = (const float*)d_in[4];
    const float* W2      = (const float*)d_in[5];
    const float* b2      = (const float*)d_in[6];
    float* out = (float*)d_out;

    const int F = 128;
    const int N = in_sizes[0] / F;
    const int E = in_sizes[1];

    // Workspace layout: [nsrc(N) | ndst(N) | buf0(N*128) | buf1(N*128)]
    char* ws = (char*)d_ws;
    float* nsrc = (float*)ws;
    float* ndst = nsrc + N;
    size_t off = ((size_t)2 * N * sizeof(float) + 255) & ~(size_t)255;
    float* buf0 = (float*)(ws + off);
    float* buf1 = buf0 + (size_t)N * F;

    const size_t featBytes = (size_t)N * F * sizeof(float);

    // --- degrees + norms ---
    hipMemsetAsync(nsrc, 0, (size_t)2 * N * sizeof(float), stream);
    gnn_deg_kernel<<<(E + 255) / 256, 256, 0, stream>>>(src, dst, nsrc, ndst, E);
    gnn_norm_kernel<<<(2 * N + 255) / 256, 256, 0, stream>>>(nsrc, 2 * N);

    const int gemmBlocks = (N + 15) / 16;
    const long long edgeThreads = (long long)E * 32;
    const int aggBlocks = (int)((edgeThreads + 255) / 256);
    const long long nodeThreads = (long long)N * 32;
    const int finBlocks = (int)((nodeThreads + 255) / 256);

    // --- layer 1: x1 = (feat*nsrc)@W1 -> buf0 ; agg -> buf1 ; h1 -> buf0 ---
    gnn_gemm_kernel<<<gemmBlocks, 128, 0, stream>>>(feature, nsrc, W1, buf0, N);
    hipMemsetAsync(buf1, 0, featBytes, stream);
    gnn_agg_kernel<<<aggBlocks, 256, 0, stream>>>(buf0, src, dst, buf1, E);
    gnn_fin_kernel<<<finBlocks, 256, 0, stream>>>(buf1, ndst, b1, buf0, N, 1);

    // --- layer 2: x2 = (h1*nsrc)@W2 -> buf1 ; agg -> buf0 ; out ---
    gnn_gemm_kernel<<<gemmBlocks, 128, 0, stream>>>(buf0, nsrc, W2, buf1, N);
    hipMemsetAsync(buf0, 0, featBytes, stream);
    gnn_agg_kernel<<<aggBlocks, 256, 0, stream>>>(buf1, src, dst, buf0, E);
    gnn_fin_kernel<<<finBlocks, 256, 0, stream>>>(buf0, ndst, b2, out, N, 0);
}